// RandomProjectLinearOperator_29171417874658
// MI455X (gfx1250) — compile-verified
//
#include <hip/hip_runtime.h>

// MI455X / gfx1250, wave32. fp32 WMMA path (V_WMMA_F32_16X16X4_F32).

typedef __attribute__((ext_vector_type(2))) float v2f;
typedef __attribute__((ext_vector_type(8))) float v8f;

#define NB    8      // batch
#define CIN_  64
#define COUT_ 64
#define NPROJ 16
#define HW_   1024   // 32*32 pixels
#define ROWS_ 512    // NB*CIN
#define K2_   1088   // NPROJ*CIN + CIN (conv fused)
#define KS2_  272    // K2_/4

#define CS_WORDS  16384            // 1024*16 coeff floats, fragment-ordered
#define TMP_STRIDE 66              // 64 + 2 pad (keeps b64 alignment, spreads banks)
#define TMP_WORDS (17*16*TMP_STRIDE)

// ---------------------------------------------------------------------------
// Pre-kernel 1: v (B,CIN,32,32) -> vs (WMMA-B fragment order) and vp (pixel major)
//   vs2[(k4*2+half)*512 + row] = { v[row][4k4+2half+0], v[row][4k4+2half+1] }
//   vp [xy*512 + row]          = v[row][xy]
// ---------------------------------------------------------------------------
__global__ __launch_bounds__(256) void rplo_swizzle_v(const float* __restrict__ v,
                                                      float* __restrict__ vs,
                                                      float* __restrict__ vp) {
  int idx = blockIdx.x * 256 + threadIdx.x;   // 131072 = 512 rows * 256 k4
  int row = idx & 511;
  int k4  = idx >> 9;                         // 0..255
  const float4 q = *(const float4*)(v + row * 1024 + k4 * 4);
  float2* vs2 = (float2*)vs;
  vs2[(k4 * 2 + 0) * 512 + row] = make_float2(q.x, q.y);
  vs2[(k4 * 2 + 1) * 512 + row] = make_float2(q.z, q.w);
  vp[(k4 * 4 + 0) * 512 + row] = q.x;
  vp[(k4 * 4 + 1) * 512 + row] = q.y;
  vp[(k4 * 4 + 2) * 512 + row] = q.z;
  vp[(k4 * 4 + 3) * 512 + row] = q.w;
}

// ---------------------------------------------------------------------------
// Pre-kernel 2: params (N,COUT,CIN) and conv_w (COUT,CIN) -> Ps fragment order.
//   K2 axis: k2<1024 -> (n=k2>>6, i=k2&63) of params ; k2>=1024 -> conv_w[:,k2-1024]
//   Ps2[(ks*2+half)*64 + o] = { B[k2][o], B[k2+1][o] },  k2 = 4*ks + 2*half
// ---------------------------------------------------------------------------
__global__ __launch_bounds__(256) void rplo_swizzle_p(const float* __restrict__ params,
                                                      const float* __restrict__ conv_w,
                                                      float* __restrict__ Ps) {
  int idx = blockIdx.x * 256 + threadIdx.x;   // 34816 = 544 * 64
  if (idx >= KS2_ * 2 * 64) return;
  int o = idx & 63;
  int g = idx >> 6;          // ks*2+half
  int k2 = g * 2;
  float x, y;
  {
    int k = k2;
    if (k < 1024) { int n = k >> 6, i = k & 63; x = params[n * 4096 + o * 64 + i]; }
    else          { x = conv_w[o * 64 + (k - 1024)]; }
  }
  {
    int k = k2 + 1;
    if (k < 1024) { int n = k >> 6, i = k & 63; y = params[n * 4096 + o * 64 + i]; }
    else          { y = conv_w[o * 64 + (k - 1024)]; }
  }
  ((float2*)Ps)[idx] = make_float2(x, y);
}

// ---------------------------------------------------------------------------
// Main fused kernel: one block per output pixel (h,w). 256 threads = 8 waves.
// ---------------------------------------------------------------------------
__global__ __launch_bounds__(256) void rplo_main(
    const float* __restrict__ vs, const float* __restrict__ vp,
    const float* __restrict__ Ps,
    const float* __restrict__ w1, const float* __restrict__ b1,
    const float* __restrict__ w2, const float* __restrict__ b2,
    const float* __restrict__ conv_b, const float* __restrict__ bias,
    float* __restrict__ out) {
  __shared__ float lds[CS_WORDS + TMP_WORDS];
  float* cs  = lds;              // coeff fragments (aliased as reduction buf later)
  float* tmp = lds + CS_WORDS;   // tmp[n 0..16][b 0..15][66] ; n==16 holds v at pixel

  const int tid  = threadIdx.x;
  const int lane = tid & 31;
  const int wav  = tid >> 5;     // 0..7
  const int lidx = lane & 15;
  const int half = lane >> 4;
  const int hw   = blockIdx.x;
  const int h    = hw >> 5, w = hw & 31;

  // ---- zero tmp (padding rows b>=8 must be zero) ----
  for (int i = tid; i < TMP_WORDS; i += 256) tmp[i] = 0.0f;
  __syncthreads();

  // ---- vloc: A-rows for the fused conv (n==16 slab) ----
  {
    const float* vrow = vp + hw * 512;
    for (int t = tid; t < 512; t += 256) {
      int b = t >> 6, i = t & 63;
      tmp[(256 + b) * TMP_STRIDE + i] = vrow[t];
    }
  }

  // ---- coeff MLP -> cs (stored directly in A-fragment order) ----
  const float chh = 2.0f * (h + 0.5f) / 32.0f - 1.0f;
  const float cww = 2.0f * (w + 0.5f) / 32.0f - 1.0f;
  for (int p = 0; p < 4; ++p) {
    int xy = p * 256 + tid;
    int x = xy >> 5, y = xy & 31;
    float cx = 2.0f * (x + 0.5f) / 32.0f - 1.0f;
    float cy = 2.0f * (y + 0.5f) / 32.0f - 1.0f;
    float hid[8];
#pragma unroll
    for (int j = 0; j < 8; ++j) {
      float z = w1[j * 4 + 0] * cx + w1[j * 4 + 1] * cy +
                w1[j * 4 + 2] * chh + w1[j * 4 + 3] * cww + b1[j];
      hid[j] = 0.5f * z * (1.0f + erff(z * 0.70710678118654752f));  // exact GELU
    }
    float* csp = cs + (xy >> 1) * 32 + (xy & 1);
#pragma unroll
    for (int n = 0; n < NPROJ; ++n) {
      float acc = b2[n];
#pragma unroll
      for (int j = 0; j < 8; ++j) acc += w2[n * 8 + j] * hid[j];
      csp[2 * n] = acc;
    }
  }
  __syncthreads();

  // ---- GEMM1: tmp[n][b*64+i] = sum_xy coeff[xy][n] * v[b*64+i][xy] ----
  v8f acc[4] = {v8f{}, v8f{}, v8f{}, v8f{}};
  {
    const v2f* cs2 = (const v2f*)cs;
    const v2f* vs2 = (const v2f*)vs;
    const v2f* aptr = cs2 + lane;                            // A frag: one b64 / k-step
    const v2f* bptr = vs2 + half * 512 + wav * 64 + lidx;    // 4 col-tiles per wave
    for (int k4 = 0; k4 < 256; ++k4) {
      v2f a  = aptr[0];
      v2f f0 = bptr[0];
      v2f f1 = bptr[16];
      v2f f2 = bptr[32];
      v2f f3 = bptr[48];
      aptr += 32;
      bptr += 1024;
      acc[0] = __builtin_amdgcn_wmma_f32_16x16x4_f32(false, a, false, f0, (short)0, acc[0], false, false);
      acc[1] = __builtin_amdgcn_wmma_f32_16x16x4_f32(false, a, false, f1, (short)0, acc[1], false, false);
      acc[2] = __builtin_amdgcn_wmma_f32_16x16x4_f32(false, a, false, f2, (short)0, acc[2], false, false);
      acc[3] = __builtin_amdgcn_wmma_f32_16x16x4_f32(false, a, false, f3, (short)0, acc[3], false, false);
    }
  }

  // ---- scatter accumulators into tmp (pre-scaled by 1/(H*W)) ----
  {
    const float inv = 1.0f / 1024.0f;
#pragma unroll
    for (int ct = 0; ct < 4; ++ct) {
      int col = wav * 64 + ct * 16 + lidx;
      int b = col >> 6, i = col & 63;
#pragma unroll
      for (int r = 0; r < 8; ++r) {
        int n = r + half * 8;
        tmp[(n * 16 + b) * TMP_STRIDE + i] = acc[ct][r] * inv;
      }
    }
  }
  __syncthreads();

  // ---- GEMM2 (conv fused): t[b][o] = sum_{k2<1088} A[b][k2] * Ps[k2][o] ----
  v8f t2 = {};
  {
    const int obase = (wav & 3) * 16;
    const int khalf = wav >> 2;
    const v2f* pb = (const v2f*)((const float2*)Ps + half * 64 + obase + lidx);
    const int ks0 = khalf * 136, ks1 = ks0 + 136;
    for (int ks = ks0; ks < ks1; ++ks) {
      int k2 = (ks << 2) + (half << 1);
      int n = k2 >> 6, i = k2 & 63;
      v2f a = *(const v2f*)(tmp + (n * 16 + lidx) * TMP_STRIDE + i);
      v2f b = pb[ks * 128];
      t2 = __builtin_amdgcn_wmma_f32_16x16x4_f32(false, a, false, b, (short)0, t2, false, false);
    }
  }

  // ---- cross-wave reduction of the two K-halves (reuse cs area) ----
  {
    float* red = cs;
#pragma unroll
    for (int r = 0; r < 8; ++r) red[wav * 256 + r * 32 + lane] = t2[r];
  }
  __syncthreads();

  // ---- epilogue: sum halves, add conv_b + bias, store ----
  {
    const float* red = cs;
    int o = tid & 63;
    int bq = tid >> 6;          // 0..3
    int ot = o >> 4, l = o & 15;
    float cb = conv_b[o] + bias[o];
#pragma unroll
    for (int q = 0; q < 2; ++q) {
      int b = bq + q * 4;       // b = 0..7
      float s = red[ot * 256 + b * 32 + l] + red[(ot + 4) * 256 + b * 32 + l];
      out[(b * 64 + o) * 1024 + hw] = s + cb;
    }
  }
}

// ---------------------------------------------------------------------------
extern "C" void kernel_launch(void* const* d_in, const int* in_sizes, int n_in,
                              void* d_out, int out_size, void* d_ws, size_t ws_size,
                              hipStream_t stream) {
  (void)in_sizes; (void)n_in; (void)out_size; (void)ws_size;
  const float* v      = (const float*)d_in[0];
  const float* params = (const float*)d_in[1];
  const float* w1     = (const float*)d_in[2];
  const float* b1     = (const float*)d_in[3];
  const float* w2     = (const float*)d_in[4];
  const float* b2     = (const float*)d_in[5];
  const float* conv_w = (const float*)d_in[6];
  const float* conv_b = (const float*)d_in[7];
  const float* bias   = (const float*)d_in[8];
  float* out = (float*)d_out;

  float* vs = (float*)d_ws;          // 524288 floats (2 MB)
  float* vp = vs + 524288;           // 524288 floats (2 MB)
  float* Ps = vp + 524288;           // 69632 floats (~272 KB)

  rplo_swizzle_v<<<512, 256, 0, stream>>>(v, vs, vp);
  rplo_swizzle_p<<<136, 256, 0, stream>>>(params, conv_w, Ps);
  rplo_main<<<1024, 256, 0, stream>>>(vs, vp, Ps, w1, b1, w2, b2, conv_b, bias, out);
}